// RetrievalTool_30855045055250
// MI455X (gfx1250) — compile-verified
//
#include <hip/hip_runtime.h>
#include <math.h>

typedef float v2f __attribute__((ext_vector_type(2)));
typedef float v8f __attribute__((ext_vector_type(8)));

#define B_    32
#define L_    512
#define C_    64
#define G_    3
#define T_    10000
#define P_    96
#define TOPM_ 20

#define WAVES_ 8
#define TILES_ (T_ / 16)   // 625

// workspace layout in floats (no more 82MB sim buffer -- fully fused)
#define WSQ_OFF   0
#define WSQ_LEN   (G_ * B_ * C_ * 4)          // 24576
#define WSW_OFF   (WSQ_OFF + WSQ_LEN)
#define WSW_LEN   (B_ * C_ * TOPM_)           // 40960
#define WSI_OFF   (WSW_OFF + WSW_LEN)

// ---------------------------------------------------------------------------
// Kernel 1: per-(b,c) streaming stats for PERIODS=[4,2,1] -> normalized q
// ---------------------------------------------------------------------------
__device__ inline void finalize_state(float s, float sq, float ad,
                                      float first, float last, float invN,
                                      float* __restrict__ dst) {
  float mp    = s * invN;            // mean of pooled series
  float mean  = mp - last;           // offset removal affects only the mean
  float var   = sq * invN - mp * mp;
  float sd    = sqrtf(fmaxf(var, 0.0f));
  float slope = (last - first) * (1.0f / (float)(L_ - 1));
  float adm   = ad * (1.0f / (float)(L_ - 1));
  float n = sqrtf(mean * mean + sd * sd + slope * slope + adm * adm);
  float inv = 1.0f / fmaxf(n, 1e-12f);
  dst[0] = mean * inv;
  dst[1] = sd * inv;
  dst[2] = slope * inv;
  dst[3] = adm * inv;
}

__global__ void qstate_kernel(const float* __restrict__ x,
                              float* __restrict__ q) {
  int tid = blockIdx.x * blockDim.x + threadIdx.x;
  if (tid >= B_ * C_) return;
  int b = tid / C_;
  int c = tid % C_;
  const float* xp = x + (size_t)b * L_ * C_ + c;

  float s1 = 0.f, q1 = 0.f, a1 = 0.f;
  float s2 = 0.f, q2 = 0.f, a2 = 0.f;
  float s4 = 0.f, q4 = 0.f, a4 = 0.f;
  float f1 = 0.f, f2 = 0.f, f4 = 0.f;
  float p1 = 0.f, p2 = 0.f, p4 = 0.f;

  for (int l0 = 0; l0 < L_; l0 += 4) {
    float v0 = xp[(l0 + 0) * C_];
    float v1 = xp[(l0 + 1) * C_];
    float v2 = xp[(l0 + 2) * C_];
    float v3 = xp[(l0 + 3) * C_];
    float e0 = 0.5f * (v0 + v1);
    float e1 = 0.5f * (v2 + v3);
    float e4 = 0.5f * (e0 + e1);

    s1 += v0 + v1 + v2 + v3;
    q1 += v0 * v0 + v1 * v1 + v2 * v2 + v3 * v3;
    s2 += e0 + e1;
    q2 += e0 * e0 + e1 * e1;
    s4 += e4;
    q4 += e4 * e4;

    if (l0 == 0) {
      f1 = v0; f2 = e0; f4 = e4;
    } else {
      a1 += fabsf(v0 - p1);
      a2 += fabsf(e0 - p2);
      a4 += fabsf(e4 - p4);
    }
    a1 += fabsf(v1 - v0) + fabsf(v2 - v1) + fabsf(v3 - v2);
    a2 += fabsf(e1 - e0);

    p1 = v3; p2 = e1; p4 = e4;
  }

  // PERIODS = [4, 2, 1] -> g = 0, 1, 2
  finalize_state(s4, q4, a4, f4, p4, 1.0f / 128.0f,
                 q + (((size_t)(0 * B_ + b) * C_ + c) << 2));
  finalize_state(s2, q2, a2, f2, p2, 1.0f / 256.0f,
                 q + (((size_t)(1 * B_ + b) * C_ + c) << 2));
  finalize_state(s1, q1, a1, f1, p1, 1.0f / 512.0f,
                 q + (((size_t)(2 * B_ + b) * C_ + c) << 2));
}

// ---------------------------------------------------------------------------
// Kernel 2 (fused): WMMA similarity + per-(b,c) top-20 + softmax.
// grid = (C_), block = 256 (8 waves). sim never leaves the WGP.
//
// Per tile: 6x v_wmma_f32_16x16x4_f32 (3 g chained into the accumulator for
// the g-mean, 2 M-tiles covering b=0..31). D-tile staged through LDS, then
// lane b maintains a register-resident sorted top-20 over its t-stream.
// Final: 8-way sorted merge + softmax by wave 0.
// ---------------------------------------------------------------------------
__global__ __launch_bounds__(WAVES_ * 32) void sim_topk_kernel(
    const float* __restrict__ q, const float* __restrict__ pool,
    float* __restrict__ wgt, int* __restrict__ idxo) {
  const int c    = blockIdx.x;
  const int tid  = threadIdx.x;
  const int wave = tid >> 5;
  const int lane = tid & 31;
  const int nl   = lane & 15;
  const int hi   = lane >> 4;          // 0 -> K{0,1}, 1 -> K{2,3}
  const int d0   = hi ? 2 : 0;

  __shared__ float tileBuf[WAVES_][B_][17];   // padded vs bank conflicts
  __shared__ float mv[WAVES_][B_][TOPM_];
  __shared__ int   mi[WAVES_][B_][TOPM_];
  __shared__ float fv[B_][TOPM_];
  __shared__ int   fi[B_][TOPM_];

  // A fragments (16x4 f32): VGPR0 = K(hi?2:0), VGPR1 = K(hi?3:1), row = lane&15
  v2f A0[G_], A1[G_];
#pragma unroll
  for (int g = 0; g < G_; ++g) {
    const float* qa = q + (((size_t)(g * B_ + nl) * C_ + c) << 2);
    const float* qb = q + (((size_t)(g * B_ + 16 + nl) * C_ + c) << 2);
    A0[g].x = qa[d0];
    A0[g].y = qa[d0 + 1];
    A1[g].x = qb[d0];
    A1[g].y = qb[d0 + 1];
  }

  // per-lane top-20 for row b = lane
  float val[TOPM_];
  int   ind[TOPM_];
#pragma unroll
  for (int j = 0; j < TOPM_; ++j) {
    val[j] = -__builtin_inff();
    ind[j] = 0;
  }

  const float4* pool4 = (const float4*)pool;

  for (int tt = wave; tt < TILES_; tt += WAVES_) {
    const int t0 = tt * 16;
    const int t  = t0 + nl;

    v2f Bf[G_];
#pragma unroll
    for (int g = 0; g < G_; ++g) {
      float4 ps = pool4[(size_t)(g * T_ + t) * C_ + c];   // L2-resident (30MB)
      float n   = sqrtf(ps.x * ps.x + ps.y * ps.y + ps.z * ps.z + ps.w * ps.w);
      float inv = 1.0f / fmaxf(n, 1e-12f);
      Bf[g].x = (hi ? ps.z : ps.x) * inv;
      Bf[g].y = (hi ? ps.w : ps.y) * inv;
    }

    v8f acc0 = {0.f, 0.f, 0.f, 0.f, 0.f, 0.f, 0.f, 0.f};
    v8f acc1 = {0.f, 0.f, 0.f, 0.f, 0.f, 0.f, 0.f, 0.f};
#pragma unroll
    for (int g = 0; g < G_; ++g) {
      acc0 = __builtin_amdgcn_wmma_f32_16x16x4_f32(
          false, A0[g], false, Bf[g], (short)0, acc0, false, false);
      acc1 = __builtin_amdgcn_wmma_f32_16x16x4_f32(
          false, A1[g], false, Bf[g], (short)0, acc1, false, false);
    }

    // stage D tile to this wave's LDS buffer (D: VGPR k -> row (hi?8:0)+k)
#pragma unroll
    for (int k = 0; k < 8; ++k) {
      int br = hi * 8 + k;
      tileBuf[wave][br][nl]      = acc0[k] * (1.0f / 3.0f);
      tileBuf[wave][16 + br][nl] = acc1[k] * (1.0f / 3.0f);
    }
    // LDS is processed in-order per wave; make the cross-lane RAW explicit.
    asm volatile("s_wait_dscnt 0x0" ::: "memory");

    // lane b = lane streams its 16 t-values into the sorted top-20
#pragma unroll
    for (int j = 0; j < 16; ++j) {
      float v = tileBuf[wave][lane][j];
      if (v > val[TOPM_ - 1]) {
        float cv = v;
        int   ci = t0 + j;
#pragma unroll
        for (int m = 0; m < TOPM_; ++m) {
          bool  gt = cv > val[m];
          float tv = val[m];
          int   ti = ind[m];
          val[m] = gt ? cv : tv;
          ind[m] = gt ? ci : ti;
          cv = gt ? tv : cv;
          ci = gt ? ti : ci;
        }
      }
    }
  }

  // publish per-wave, per-b sorted lists
#pragma unroll
  for (int j = 0; j < TOPM_; ++j) {
    mv[wave][lane][j] = val[j];
    mi[wave][lane][j] = ind[j];
  }
  __syncthreads();

  // wave 0: lane b merges WAVES_ sorted lists, then fused softmax
  if (tid < B_) {
    const int b = tid;
    int p[WAVES_];
#pragma unroll
    for (int w = 0; w < WAVES_; ++w) p[w] = 0;

    float maxv = 0.f, sum = 0.f;
    for (int m = 0; m < TOPM_; ++m) {
      float best = -__builtin_inff();
      int   bidx = 0;
      int   bw   = 0;
#pragma unroll
      for (int w = 0; w < WAVES_; ++w) {
        float v = (p[w] < TOPM_) ? mv[w][b][p[w]] : -__builtin_inff();
        int   i = (p[w] < TOPM_) ? mi[w][b][p[w]] : 0;
        if (v > best) { best = v; bidx = i; bw = w; }
      }
#pragma unroll
      for (int w = 0; w < WAVES_; ++w)
        if (w == bw) p[w]++;

      fv[b][m] = best;
      fi[b][m] = bidx;
      if (m == 0) maxv = best;            // descending order -> first is max
      sum += __expf((best - maxv) * 10.0f);
    }

    float inv = 1.0f / sum;
    for (int m = 0; m < TOPM_; ++m) {
      float w = __expf((fv[b][m] - maxv) * 10.0f) * inv;
      wgt[((size_t)b * C_ + c) * TOPM_ + m]  = w;
      idxo[((size_t)b * C_ + c) * TOPM_ + m] = fi[b][m];
    }
  }
}

// ---------------------------------------------------------------------------
// Kernel 3: out[b,p,c] = sum_m w[b,c,m] * y_pool[idx[b,c,m], p, c]
// grid = (B_, P_/4), block = 256 (= 4 p x 64 c, c fastest -> coalesced gathers)
// ---------------------------------------------------------------------------
__global__ void gather_combine_kernel(const float* __restrict__ y_pool,
                                      const float* __restrict__ w,
                                      const int* __restrict__ idx,
                                      float* __restrict__ out) {
  int b = blockIdx.x;
  int c = threadIdx.x & (C_ - 1);
  int p = blockIdx.y * 4 + (threadIdx.x >> 6);

  const float* wp = w + ((size_t)b * C_ + c) * TOPM_;
  const int*   ip = idx + ((size_t)b * C_ + c) * TOPM_;

  float acc = 0.f;
#pragma unroll
  for (int m = 0; m < TOPM_; ++m) {
    int t = ip[m];
    acc += wp[m] * y_pool[((size_t)t * P_ + p) * C_ + c];
  }
  out[((size_t)b * P_ + p) * C_ + c] = acc;
}

// ---------------------------------------------------------------------------
extern "C" void kernel_launch(void* const* d_in, const int* in_sizes, int n_in,
                              void* d_out, int out_size, void* d_ws,
                              size_t ws_size, hipStream_t stream) {
  const float* x     = (const float*)d_in[0];  // [32,512,64]
  const float* pool  = (const float*)d_in[1];  // [3,10000,64,4]
  const float* ypool = (const float*)d_in[2];  // [10000,96,64]
  float* out = (float*)d_out;                  // [32,96,64]

  float* wsf = (float*)d_ws;
  float* q   = wsf + WSQ_OFF;
  float* wgt = wsf + WSW_OFF;
  int*   idx = (int*)(wsf + WSI_OFF);

  // 1) q-state stats + L2 normalize
  qstate_kernel<<<dim3((B_ * C_ + 255) / 256), dim3(256), 0, stream>>>(x, q);

  // 2) fused WMMA similarity + top-20 + softmax (sim never hits HBM)
  sim_topk_kernel<<<dim3(C_), dim3(WAVES_ * 32), 0, stream>>>(q, pool, wgt,
                                                              idx);

  // 3) gather + weighted combine
  gather_combine_kernel<<<dim3(B_, P_ / 4), dim3(256), 0, stream>>>(ypool, wgt,
                                                                    idx, out);
}